// IntegrateWithWeightsModule_51780125721339
// MI455X (gfx1250) — compile-verified
//
#include <hip/hip_runtime.h>
#include <stdint.h>

// ---------------------------------------------------------------------------
// Segmented weighted RGB reduction (NeRF ray integration) for gfx1250.
// Bandwidth-bound: 168 MB in / 1.5 MB out -> async global->LDS staging path.
// ---------------------------------------------------------------------------

#define BLOCK 256
#define SPT   32                       // samples per thread (contiguous strip)
#define TILE  (BLOCK * SPT)            // 8192 samples per workgroup
#define STRIP_B      (SPT * 12)        // 384 bytes of rgb per thread strip
#define STRIP_STRIDE (STRIP_B + 8)     // 392 B = 98 dwords; gcd(98,64)=2 ->
                                       // 32 lanes hit 32 distinct LDS banks,
                                       // and 8B alignment for b64 chunks holds
#define LDS_BYTES (BLOCK * STRIP_STRIDE)           // 100352 B (<320KB, 3 WGs/WGP)
#define CHUNKS_PER_THREAD ((TILE * 12) / 8 / BLOCK) // 48 async b64 per thread

#define AS1 __attribute__((address_space(1)))
#define AS3 __attribute__((address_space(3)))

typedef int v2i __attribute__((ext_vector_type(2)));

__device__ __forceinline__ void async_copy_b64(const void* g, void* l) {
#if __has_builtin(__builtin_amdgcn_global_load_async_to_lds_b64)
  __builtin_amdgcn_global_load_async_to_lds_b64(
      (AS1 v2i*)(uintptr_t)g, (AS3 v2i*)l, 0, 0);
#else
  unsigned loff = (unsigned)(uintptr_t)(AS3 char*)l;
  unsigned long long ga = (unsigned long long)(uintptr_t)g;
  asm volatile("global_load_async_to_lds_b64 %0, %1, off"
               :: "v"(loff), "v"(ga) : "memory");
#endif
}

__device__ __forceinline__ void wait_async_zero() {
#if __has_builtin(__builtin_amdgcn_s_wait_asynccnt)
  __builtin_amdgcn_s_wait_asynccnt(0);
#else
  asm volatile("s_wait_asynccnt 0" ::: "memory");
#endif
}

__device__ __forceinline__ void atomic_fadd(float* p, float v) {
  __hip_atomic_fetch_add(p, v, __ATOMIC_RELAXED, __HIP_MEMORY_SCOPE_AGENT);
}

__global__ void __launch_bounds__(BLOCK) nerf_integrate_kernel(
    const int*   __restrict__ seg,
    const float* __restrict__ rgb,
    const float* __restrict__ wgt,
    float*       __restrict__ out,
    int n)
{
  extern __shared__ char smem[];
  const int  t    = threadIdx.x;
  const long tile = (long)blockIdx.x * TILE;

  // ---- Stage the tile's rgb into LDS with coalesced async b64 copies. ----
  // Chunk c covers tile bytes [c*8, c*8+8); lanes take consecutive chunks so
  // each wave instruction moves 256 contiguous bytes. LDS destination is the
  // padded per-thread-strip layout (chunks never straddle a 384B strip).
  const char* gsrc  = (const char*)rgb + tile * 12;
  const long  avail = (long)n * 12 - tile * 12;
  if (avail >= (long)TILE * 12) {
#pragma unroll
    for (int i = 0; i < CHUNKS_PER_THREAD; ++i) {
      int b = (t + i * BLOCK) * 8;
      int strip  = b / STRIP_B;
      int within = b - strip * STRIP_B;
      async_copy_b64(gsrc + b, smem + strip * STRIP_STRIDE + within);
    }
  } else {
#pragma unroll
    for (int i = 0; i < CHUNKS_PER_THREAD; ++i) {
      int b = (t + i * BLOCK) * 8;
      if ((long)b + 8 <= avail) {
        int strip  = b / STRIP_B;
        int within = b - strip * STRIP_B;
        async_copy_b64(gsrc + b, smem + strip * STRIP_STRIDE + within);
      }
    }
  }
  wait_async_zero();     // this wave's LDS writes done
  __syncthreads();       // all waves' LDS writes visible

  // ---- Each thread scans its contiguous 32-sample strip. ----
  const long base = tile + (long)t * SPT;
  if (base >= n) return;
  const int cnt = (n - base >= SPT) ? SPT : (int)(n - base);

  AS3 const float* lr =
      (AS3 const float*)(AS3 const char*)(smem + t * STRIP_STRIDE);
  const int4*   id4p = (const int4*)(seg + base);   // base % 32 == 0 -> aligned
  const float4* w4p  = (const float4*)(wgt + base);

  float a0 = 0.f, a1 = 0.f, a2 = 0.f;
  int cur = -1;

  if (cnt == SPT) {
    // Fast path: full strip, no per-sample bounds checks.
#pragma unroll
    for (int i = 0; i < SPT / 4; ++i) {
      int4   id4 = id4p[i];
      float4 w4  = w4p[i];
#pragma unroll
      for (int j = 0; j < 4; ++j) {
        int   s  = i * 4 + j;
        int   id = (j == 0) ? id4.x : (j == 1) ? id4.y : (j == 2) ? id4.z : id4.w;
        float ww = (j == 0) ? w4.x  : (j == 1) ? w4.y  : (j == 2) ? w4.z  : w4.w;
        float r0 = lr[s * 3 + 0] * ww;
        float r1 = lr[s * 3 + 1] * ww;
        float r2 = lr[s * 3 + 2] * ww;
        if (id != cur) {
          if (cur >= 0) {
            atomic_fadd(out + 3 * cur + 0, a0);
            atomic_fadd(out + 3 * cur + 1, a1);
            atomic_fadd(out + 3 * cur + 2, a2);
          }
          cur = id; a0 = r0; a1 = r1; a2 = r2;
        } else {
          a0 += r0; a1 += r1; a2 += r2;
        }
      }
    }
  } else {
    // Tail path: partial strip, scalar guarded loop.
    for (int s = 0; s < cnt; ++s) {
      int   id = seg[base + s];
      float ww = wgt[base + s];
      float r0 = lr[s * 3 + 0] * ww;
      float r1 = lr[s * 3 + 1] * ww;
      float r2 = lr[s * 3 + 2] * ww;
      if (id != cur) {
        if (cur >= 0) {
          atomic_fadd(out + 3 * cur + 0, a0);
          atomic_fadd(out + 3 * cur + 1, a1);
          atomic_fadd(out + 3 * cur + 2, a2);
        }
        cur = id; a0 = r0; a1 = r1; a2 = r2;
      } else {
        a0 += r0; a1 += r1; a2 += r2;
      }
    }
  }
  if (cur >= 0) {
    atomic_fadd(out + 3 * cur + 0, a0);
    atomic_fadd(out + 3 * cur + 1, a1);
    atomic_fadd(out + 3 * cur + 2, a2);
  }
}

__global__ void __launch_bounds__(BLOCK) nerf_zero_out(float* out, int m) {
  int i = blockIdx.x * blockDim.x + threadIdx.x;
  if (i < m) out[i] = 0.0f;
}

extern "C" void kernel_launch(void* const* d_in, const int* in_sizes, int n_in,
                              void* d_out, int out_size, void* d_ws, size_t ws_size,
                              hipStream_t stream) {
  const int*   seg = (const int*)d_in[0];    // [N] int32 sorted segment ids
  const float* rgb = (const float*)d_in[1];  // [N,3] f32
  const float* wgt = (const float*)d_in[2];  // [N,1] f32
  float*       out = (float*)d_out;          // [n_rays,3] f32
  const int n = in_sizes[0];
  const int m = out_size;                    // n_rays * 3

  hipLaunchKernelGGL(nerf_zero_out, dim3((m + BLOCK - 1) / BLOCK), dim3(BLOCK),
                     0, stream, out, m);

  const int grid = (n + TILE - 1) / TILE;
  hipLaunchKernelGGL(nerf_integrate_kernel, dim3(grid), dim3(BLOCK),
                     LDS_BYTES, stream, seg, rgb, wgt, out, n);
}